// Rasterizer_28003186770311
// MI455X (gfx1250) — compile-verified
//
#include <hip/hip_runtime.h>
#include <hip/hip_bf16.h>
#include <stdint.h>

// Rasterizer post-process for MI455X (gfx1250).
// Memory-bound (~380 MB HBM traffic -> ~16us floor @ 23.3 TB/s).
// Uses CDNA5 async global->LDS copies (ASYNCcnt-tracked) to stage zbuf tiles.
// pixel_kernel is defined first so the disasm snippet shows the async ops.

#define NMESH 8
#define CCH   32
#define BIG_Z 1000000.0f

// ---------------------------------------------------------------------------
// Kernel B (defined first for disasm visibility): fused per-pixel pass.
// One thread per (h,w), loops over 8 meshes. Stages zbuf[n][pix] for the
// block's 256 pixels into LDS with CDNA5 async global->LDS copies, then
// resolves duplicates and emits all per-pixel outputs with coalesced stores.
// ---------------------------------------------------------------------------
__global__ __launch_bounds__(256) void pixel_kernel(
    const float* __restrict__ zbuf,      // (N,H,W,1)
    const int*   __restrict__ ptf,       // (N,H,W,1)
    const float* __restrict__ bary,      // (N,H,W,1,3)
    const float* __restrict__ dists,     // (N,H,W,1)
    const float* __restrict__ fmem,      // (F,3,C)
    const int*   __restrict__ maskdup_p, // scalar
    float* __restrict__ out_map,         // (N,C,H,W)
    float* __restrict__ out_mask,        // (N,H,W)
    float* __restrict__ out_zb,          // (N,H,W,1)
    float* __restrict__ out_p,           // (N,H,W,1)  (int values as float)
    float* __restrict__ out_d,           // (N,H,W,1)
    float* __restrict__ out_bc,          // (N,H,W,1,3)
    unsigned* __restrict__ flags,        // (F) face-hit flags
    int HW)
{
    __shared__ float s_z[NMESH * 256];

    const int tid = threadIdx.x;
    const int pix = blockIdx.x * 256 + tid;
    if (pix >= HW) return;

    // ---- CDNA5 async global->LDS staging of the 8 z-planes for this tile ----
    // LDS byte offset = low 32 bits of the generic pointer to shared memory.
    // NOTE: offset: immediate is added to BOTH lds and global addresses for
    // async-to-LDS ops (ISA 15.18), so per-plane addressing stays in VGPRs.
    uint32_t lds_base = (uint32_t)(uintptr_t)(&s_z[0]);
#pragma unroll
    for (int n = 0; n < NMESH; ++n) {
        uint32_t la = lds_base + (uint32_t)(((n << 8) + tid) << 2);
        uint64_t ga = (uint64_t)(uintptr_t)(zbuf + (size_t)n * HW + pix);
        asm volatile("global_load_async_to_lds_b32 %0, %1, off"
                     :: "v"(la), "v"(ga) : "memory");
    }
    asm volatile("s_wait_asynccnt 0" ::: "memory");

    float z[NMESH];
#pragma unroll
    for (int n = 0; n < NMESH; ++n) z[n] = s_z[(n << 8) + tid];

    // ---- duplicate resolution across meshes ----
    int   cover  = 0;
    int   minIdx = 0;
    float minz   = BIG_Z + 1.0f;
#pragma unroll
    for (int n = 0; n < NMESH; ++n) {
        cover += (z[n] > -1.0f) ? 1 : 0;
        float t = (z[n] < 0.0f) ? BIG_Z : z[n];
        if (t < minz) { minz = t; minIdx = n; }   // strict < => first occurrence (argmin tie rule)
    }
    const int maskdup = *maskdup_p;

    for (int n = 0; n < NMESH; ++n) {
        const size_t gidx = (size_t)n * HW + pix;
        const bool dup = (cover > 1) && (maskdup != 0 || n != minIdx);

        const float zorig = z[n];
        const int   p     = dup ? -1 : ptf[gidx];
        const float dd    = dup ? -1.0f : dists[gidx];
        float b0, b1, b2;
        if (dup) {
            b0 = b1 = b2 = -1.0f;
        } else {
            b0 = bary[gidx * 3 + 0];
            b1 = bary[gidx * 3 + 1];
            b2 = bary[gidx * 3 + 2];
        }

        out_mask[gidx]       = (zorig >= 0.0f) ? 1.0f : 0.0f;  // per_obj_mask uses ORIGINAL zbuf
        out_zb[gidx]         = dup ? -1.0f : zorig;
        out_p[gidx]          = (float)p;
        out_d[gidx]          = dd;
        out_bc[gidx * 3 + 0] = b0;
        out_bc[gidx * 3 + 1] = b1;
        out_bc[gidx * 3 + 2] = b2;

        const bool valid = (p >= 0);
        float* omap = out_map + ((size_t)n * CCH) * HW + pix;

        if (valid) {
            flags[p] = 1u;  // idempotent flag store (all writers store 1u) — no atomic needed
            const float4* fa = (const float4*)(fmem + (size_t)p * (3 * CCH));
#pragma unroll
            for (int c4 = 0; c4 < CCH / 4; ++c4) {
                float4 a0 = fa[c4];
                float4 a1 = fa[(CCH / 4) + c4];
                float4 a2 = fa[2 * (CCH / 4) + c4];
                float o0 = b0 * a0.x + b1 * a1.x + b2 * a2.x;
                float o1 = b0 * a0.y + b1 * a1.y + b2 * a2.y;
                float o2 = b0 * a0.z + b1 * a1.z + b2 * a2.z;
                float o3 = b0 * a0.w + b1 * a1.w + b2 * a2.w;
                // NCHW: stride HW between channels; lanes are consecutive w -> 128B bursts per store
                omap[(size_t)(c4 * 4 + 0) * HW] = o0;
                omap[(size_t)(c4 * 4 + 1) * HW] = o1;
                omap[(size_t)(c4 * 4 + 2) * HW] = o2;
                omap[(size_t)(c4 * 4 + 3) * HW] = o3;
            }
        } else {
#pragma unroll
            for (int c = 0; c < CCH; ++c) omap[(size_t)c * HW] = 0.0f;
        }
    }
}

// ---------------------------------------------------------------------------
// Kernel A: zero vertex-visibility output region and face-flag scratch.
// ---------------------------------------------------------------------------
__global__ __launch_bounds__(256) void zero_kernel(float* __restrict__ vv, int nv,
                                                   unsigned* __restrict__ flags, int F) {
    int i = blockIdx.x * 256 + threadIdx.x;
    if (i < nv) vv[i] = 0.0f;
    if (i < F)  flags[i] = 0u;
}

// ---------------------------------------------------------------------------
// Kernel C: face visibility -> vertex visibility. Idempotent 1.0f stores.
// ---------------------------------------------------------------------------
__global__ __launch_bounds__(256) void vertex_kernel(const unsigned* __restrict__ flags,
                                                     const int* __restrict__ packed_faces,
                                                     float* __restrict__ vv, int F) {
    int f = blockIdx.x * 256 + threadIdx.x;
    if (f < F && flags[f] != 0u) {
        vv[packed_faces[3 * f + 0]] = 1.0f;
        vv[packed_faces[3 * f + 1]] = 1.0f;
        vv[packed_faces[3 * f + 2]] = 1.0f;
    }
}

extern "C" void kernel_launch(void* const* d_in, const int* in_sizes, int n_in,
                              void* d_out, int out_size, void* d_ws, size_t ws_size,
                              hipStream_t stream) {
    (void)n_in; (void)ws_size;

    const float* zbuf  = (const float*)d_in[0];
    const int*   ptf   = (const int*)  d_in[1];
    const float* bary  = (const float*)d_in[2];
    const float* dists = (const float*)d_in[3];
    const float* fmem  = (const float*)d_in[4];
    const int*   pfcs  = (const int*)  d_in[5];
    // d_in[6] = num_verts (unused on device; region size derived below)
    const int*   mdup  = (const int*)  d_in[7];

    const int NHW = in_sizes[0];        // N*H*W*K = 8*512*512
    const int HW  = NHW / NMESH;        // 512*512
    const int F   = in_sizes[5] / 3;    // 20000
    // Output layout (concatenated, element offsets):
    //   out_map (NHW*C) | vv (nv) | per_obj_mask (NHW) | zb (NHW) | p (NHW) | d (NHW) | bc (NHW*3)
    const long long nonvv = (long long)NHW * CCH + (long long)NHW * 7; // everything except vv
    const int nv = (int)((long long)out_size - nonvv);                 // = num_verts

    float* base    = (float*)d_out;
    float* o_map   = base;
    float* o_vv    = o_map + (size_t)NHW * CCH;
    float* o_mask  = o_vv + nv;
    float* o_zb    = o_mask + NHW;
    float* o_p     = o_zb + NHW;
    float* o_d     = o_p + NHW;
    float* o_bc    = o_d + NHW;

    unsigned* flags = (unsigned*)d_ws;  // F * 4 bytes of scratch

    const int zn = (nv > F ? nv : F);
    zero_kernel<<<(zn + 255) / 256, 256, 0, stream>>>(o_vv, nv, flags, F);

    pixel_kernel<<<(HW + 255) / 256, 256, 0, stream>>>(
        zbuf, ptf, bary, dists, fmem, mdup,
        o_map, o_mask, o_zb, o_p, o_d, o_bc, flags, HW);

    vertex_kernel<<<(F + 255) / 256, 256, 0, stream>>>(flags, pfcs, o_vv, F);
}